// custom_GINConv_71863392796754
// MI455X (gfx1250) — compile-verified
//
#include <hip/hip_runtime.h>
#include <math.h>

// ---------------------------------------------------------------------------
// GIN+GAT fused graph conv for MI455X (gfx1250, wave32).
//   rst = fd1 + segsum(fs1[src]) + segsum(softmax_dst(leaky(el[src]+er[dst])) * fs2[src])
// GEMMs run on the fp32 WMMA pipe (v_wmma_f32_16x16x4_f32) — exact fp32 MACs,
// matching the fp32 reference. Edge phase is HBM-bound scatter/gather.
// ---------------------------------------------------------------------------

typedef __attribute__((ext_vector_type(2))) float v2f;
typedef __attribute__((ext_vector_type(8))) float v8f;

#define DFEAT 128
#define NEG_SLOPE 0.2f

// order-preserving float<->uint encoding for atomic max over signed floats
__device__ __forceinline__ unsigned f2ord(float f) {
  unsigned u = __float_as_uint(f);
  return (u & 0x80000000u) ? ~u : (u | 0x80000000u);
}
__device__ __forceinline__ float ord2f(unsigned k) {
  return (k & 0x80000000u) ? __uint_as_float(k & 0x7fffffffu)
                           : __uint_as_float(~k);
}

// ---------------------------------------------------------------------------
// Kernel 0: init softmax state + fold attn vectors through W_src2/W_dst2:
//   wl = attn_l @ W_src2   (so el = feat @ wl), wr = attn_r @ W_dst2
// This eliminates the fd2 GEMM entirely.
// ---------------------------------------------------------------------------
__global__ void prep_kernel(const float* __restrict__ W_src2,
                            const float* __restrict__ W_dst2,
                            const float* __restrict__ attn_l,
                            const float* __restrict__ attn_r,
                            float* __restrict__ wl, float* __restrict__ wr,
                            unsigned* __restrict__ seg_key,
                            float* __restrict__ denom, int n) {
  int idx = blockIdx.x * blockDim.x + threadIdx.x;
  if (idx < n) { seg_key[idx] = 0u; denom[idx] = 0.0f; }  // 0u == ord(-inf)-eps
  if (idx < DFEAT) {
    float sl = 0.f, sr = 0.f;
    for (int j = 0; j < DFEAT; ++j) {
      sl += attn_l[j] * W_src2[j * DFEAT + idx];
      sr += attn_r[j] * W_dst2[j * DFEAT + idx];
    }
    wl[idx] = sl; wr[idx] = sr;
  }
}

// ---------------------------------------------------------------------------
// Kernel 1: three GEMMs out = feat @ W.T via fp32 WMMA 16x16x4.
// Block = 256 threads = 8 waves; block owns one 16-row M tile, stages the
// 16x128 fp32 A tile in LDS once, each wave computes one 16-col N tile.
// grid = (n/16, 3): y selects {W_src->fs1, W_dst->rst(=fd1), W_src2->fs2}.
// ---------------------------------------------------------------------------
__global__ void __launch_bounds__(256)
gemm3_wmma_kernel(const float* __restrict__ feat,
                  const float* __restrict__ W_src,
                  const float* __restrict__ W_dst,
                  const float* __restrict__ W_src2,
                  float* __restrict__ fs1,
                  float* __restrict__ out_fd1,
                  float* __restrict__ fs2) {
  __shared__ float Atile[16 * DFEAT];  // 8 KB of the 320 KB WGP LDS

  const int mtile = blockIdx.x;
  const int which = blockIdx.y;
  const float* __restrict__ W = (which == 0) ? W_src : (which == 1) ? W_dst : W_src2;
  float* __restrict__ Out = (which == 0) ? fs1 : (which == 1) ? out_fd1 : fs2;
  const int mbase = mtile * 16;

  // cooperative 16x128 fp32 tile load: 512 float4s / 256 threads
  {
    const float4* s4 = (const float4*)(feat + (size_t)mbase * DFEAT);
    float4* d4 = (float4*)Atile;
    for (int i = threadIdx.x; i < 16 * DFEAT / 4; i += blockDim.x) d4[i] = s4[i];
  }
  __syncthreads();

  const int lane  = threadIdx.x & 31;
  const int wv    = threadIdx.x >> 5;   // 0..7 -> N tile
  const int nbase = wv * 16;
  const int mrow  = lane & 15;          // A row (M) / C column (N) index
  const int khalf = (lane >> 4) * 2;    // A/B K sub-offset: lanes 0-15 -> K{0,1}, 16-31 -> K{2,3}
  const int col   = nbase + mrow;       // output column held by this lane

  v8f acc = {};
  #pragma unroll 8
  for (int kk = 0; kk < DFEAT / 4; ++kk) {
    const int kb = kk * 4 + khalf;      // even -> 8B-aligned ds_load_b64
    v2f a = *(const v2f*)&Atile[mrow * DFEAT + kb];        // A[M=mrow, K=kb..kb+1]
    v2f b = *(const v2f*)&W[(size_t)col * DFEAT + kb];     // B[K,N=col] = W[col,K]
    acc = __builtin_amdgcn_wmma_f32_16x16x4_f32(
        /*neg_a=*/false, a, /*neg_b=*/false, b,
        /*c_mod=*/(short)0, acc, /*reuse_a=*/false, /*reuse_b=*/false);
  }

  // C/D layout: VGPR v -> row v + 8*(lane>>4), col = lane&15 (+nbase)
  const int rbase = (lane >> 4) * 8;
  float* __restrict__ outp = Out + (size_t)mbase * DFEAT + col;
  #pragma unroll
  for (int v = 0; v < 8; ++v) outp[(size_t)(rbase + v) * DFEAT] = acc[v];
}

// ---------------------------------------------------------------------------
// Kernel 2: per-node attention logits el = feat.wl, er = feat.wr (streaming)
// ---------------------------------------------------------------------------
__global__ void elr_kernel(const float* __restrict__ feat,
                           const float* __restrict__ wl,
                           const float* __restrict__ wr,
                           float* __restrict__ el, float* __restrict__ er, int n) {
  int i = blockIdx.x * blockDim.x + threadIdx.x;
  if (i >= n) return;
  const float4* f4 = (const float4*)(feat + (size_t)i * DFEAT);
  const float4* l4 = (const float4*)wl;
  const float4* r4 = (const float4*)wr;
  float sl = 0.f, sr = 0.f;
  #pragma unroll 8
  for (int k = 0; k < DFEAT / 4; ++k) {
    float4 f = f4[k], l = l4[k], r = r4[k];
    sl += f.x * l.x + f.y * l.y + f.z * l.z + f.w * l.w;
    sr += f.x * r.x + f.y * r.y + f.z * r.z + f.w * r.w;
  }
  el[i] = sl; er[i] = sr;
}

__device__ __forceinline__ float edge_logit(const float* el, const float* er,
                                            int s, int d) {
  float e = el[s] + er[d];
  return (e >= 0.f) ? e : NEG_SLOPE * e;
}

// Kernel 3: segment max over dst (ordered-uint atomicMax)
__global__ void edge_max_kernel(const int* __restrict__ src,
                                const int* __restrict__ dst,
                                const float* __restrict__ el,
                                const float* __restrict__ er,
                                unsigned* __restrict__ seg_key, int nE) {
  int i = blockIdx.x * blockDim.x + threadIdx.x;
  if (i >= nE) return;
  int d = dst[i];
  atomicMax(&seg_key[d], f2ord(edge_logit(el, er, src[i], d)));
}

// Kernel 4: softmax denominator
__global__ void edge_sum_kernel(const int* __restrict__ src,
                                const int* __restrict__ dst,
                                const float* __restrict__ el,
                                const float* __restrict__ er,
                                const unsigned* __restrict__ seg_key,
                                float* __restrict__ denom, int nE) {
  int i = blockIdx.x * blockDim.x + threadIdx.x;
  if (i >= nE) return;
  int d = dst[i];
  float e = edge_logit(el, er, src[i], d);
  atomicAdd(&denom[d], __expf(e - ord2f(seg_key[d])));
}

// ---------------------------------------------------------------------------
// Kernel 5: one wave per edge; out[dst] += fs1[src] + a * fs2[src].
// 4 features per lane (float4 gathers) -> HBM-bound, dominates runtime.
// ---------------------------------------------------------------------------
__global__ void __launch_bounds__(256)
scatter_kernel(const int* __restrict__ src, const int* __restrict__ dst,
               const float* __restrict__ el, const float* __restrict__ er,
               const unsigned* __restrict__ seg_key,
               const float* __restrict__ denom,
               const float* __restrict__ fs1, const float* __restrict__ fs2,
               float* __restrict__ out, int nE) {
  int gw = (int)((blockIdx.x * (unsigned)blockDim.x + threadIdx.x) >> 5);
  int lane = threadIdx.x & 31;
  if (gw >= nE) return;
  int s = src[gw], d = dst[gw];
  float e = edge_logit(el, er, s, d);
  float a = __expf(e - ord2f(seg_key[d])) / denom[d];
  int f = lane * 4;
  float4 v1 = *(const float4*)(fs1 + (size_t)s * DFEAT + f);
  float4 v2 = *(const float4*)(fs2 + (size_t)s * DFEAT + f);
  float* o = out + (size_t)d * DFEAT + f;
  atomicAdd(o + 0, v1.x + a * v2.x);
  atomicAdd(o + 1, v1.y + a * v2.y);
  atomicAdd(o + 2, v1.z + a * v2.z);
  atomicAdd(o + 3, v1.w + a * v2.w);
}

// ---------------------------------------------------------------------------
extern "C" void kernel_launch(void* const* d_in, const int* in_sizes, int n_in,
                              void* d_out, int out_size, void* d_ws, size_t ws_size,
                              hipStream_t stream) {
  const float* feat   = (const float*)d_in[0];
  const float* W_src  = (const float*)d_in[1];
  const float* W_dst  = (const float*)d_in[2];
  const float* W_src2 = (const float*)d_in[3];
  const float* W_dst2 = (const float*)d_in[4];
  const float* attn_l = (const float*)d_in[5];
  const float* attn_r = (const float*)d_in[6];
  const int*   src    = (const int*)d_in[7];
  const int*   dst    = (const int*)d_in[8];
  float* out = (float*)d_out;

  const int n  = in_sizes[0] / DFEAT;   // 40000 nodes
  const int nE = in_sizes[7];           // 640000 edges
  const size_t ND = (size_t)n * DFEAT;

  // workspace layout (floats): fs1 | fs2 | el | er | wl | wr | denom | seg_key
  float* ws = (float*)d_ws;
  float*    fs1     = ws;
  float*    fs2     = fs1 + ND;
  float*    el      = fs2 + ND;
  float*    er      = el + n;
  float*    wl      = er + n;
  float*    wr      = wl + DFEAT;
  float*    denom   = wr + DFEAT;
  unsigned* seg_key = (unsigned*)(denom + n);

  // 0) init + attn-folded weight vectors
  prep_kernel<<<(n + 255) / 256, 256, 0, stream>>>(
      W_src2, W_dst2, attn_l, attn_r, wl, wr, seg_key, denom, n);

  // 1) three fp32 WMMA GEMMs: fs1, fd1 (written straight into out), fs2
  gemm3_wmma_kernel<<<dim3(n / 16, 3), 256, 0, stream>>>(
      feat, W_src, W_dst, W_src2, fs1, out, fs2);

  // 2) attention logits
  elr_kernel<<<(n + 255) / 256, 256, 0, stream>>>(feat, wl, wr, el, er, n);

  // 3) segment max, 4) segment exp-sum
  edge_max_kernel<<<(nE + 255) / 256, 256, 0, stream>>>(src, dst, el, er, seg_key, nE);
  edge_sum_kernel<<<(nE + 255) / 256, 256, 0, stream>>>(src, dst, el, er, seg_key, denom, nE);

  // 5) fused neighbor-sum + attention-weighted scatter (one wave per edge)
  scatter_kernel<<<((size_t)nE * 32 + 255) / 256, 256, 0, stream>>>(
      src, dst, el, er, seg_key, denom, fs1, fs2, out, nE);
}